// DenseGATLayer_15891378995371
// MI455X (gfx1250) — compile-verified
//
#include <hip/hip_runtime.h>
#include <math.h>

// Problem constants (from reference): B=2, N=8192, D=256, P=3, HEADS=4, hd=64, K=16
#define B_   2
#define N_   8192
#define D_   256
#define H_   4
#define HD_  64
#define K_   16
#define SEG  8            // candidate segments for KNN parallelism
#define SEGLEN (N_/SEG)   // 1024
#define TQ   256          // queries per block (knn)
#define TC   256          // candidate chunk staged in LDS (768 floats = 3 KB)
#define NCHUNK (SEGLEN/TC)

typedef __attribute__((ext_vector_type(2))) float v2f;
typedef __attribute__((ext_vector_type(8))) float v8f;
typedef __attribute__((ext_vector_type(4))) unsigned int u32x4;
typedef __attribute__((ext_vector_type(8))) int i32x8;
typedef __attribute__((ext_vector_type(4))) int i32x4;

// ---------------------------------------------------------------------------
// TDM: 1-D tensor_load_to_lds of nelem f32 from `gaddr` to LDS byte offset
// `lds_off`. D# per CDNA5 ISA §8:
//   group0: [1:0]=count=1, [63:32]=lds_addr, [120:64]=global_addr, [127:126]=2
//   group1: [17:16]=data_size(4B), tensor_dim0=tile_dim0=nelem, strides=0
//   groups 2/3: zero (<=2-D tensor)
// Tracked by TENSORcnt -> __builtin_amdgcn_s_wait_tensorcnt.
// ---------------------------------------------------------------------------
__device__ __forceinline__ void tdm_load_1d(unsigned lds_off, const void* gaddr,
                                            int nelem) {
  const unsigned long long ga = (unsigned long long)(uintptr_t)gaddr;
  u32x4 g0 = {};
  g0[0] = 1u;                                      // count=1, user mode
  g0[1] = lds_off;                                 // lds_addr (bytes)
  g0[2] = (unsigned)ga;                            // global_addr[31:0]
  g0[3] = (unsigned)(ga >> 32) | (2u << 30);       // global_addr[56:32] | type=2
  i32x8 g1 = {};
  g1[0] = (2 << 16);                               // data_size=4B, wg_mask=0
  g1[1] = (nelem & 0xFFFF) << 16;                  // tensor_dim0[15:0]
  g1[2] = (nelem >> 16) & 0xFFFF;                  // tensor_dim0[31:16]
  g1[3] = (nelem & 0xFFFF) << 16;                  // tile_dim0
  i32x4 z4 = {};
#if __clang_major__ >= 23
  i32x8 z8 = {};
  __builtin_amdgcn_tensor_load_to_lds(g0, g1, z4, z4, z8, 0);
#else
  __builtin_amdgcn_tensor_load_to_lds(g0, g1, z4, z4, 0);
#endif
}

// ---------------------------------------------------------------------------
// Phase 1 KNN: per (query, segment) partial top-16 by squared distance.
// One thread per query. Candidate tiles are double-buffered in LDS and staged
// by the Tensor Data Mover (wave 0 issues the DMA for tile c+1 while all 8
// waves compute distances on tile c), so DMA overlaps the VALU-bound top-k.
// ---------------------------------------------------------------------------
__global__ __launch_bounds__(256) void knn_part(const float* __restrict__ pos,
                                                float* __restrict__ pd,
                                                int*   __restrict__ pi) {
  const int tid  = threadIdx.x;
  const int wave = tid >> 5;
  const int n    = blockIdx.x * TQ + tid;
  const int seg  = blockIdx.y;
  const int b    = blockIdx.z;
  const float* pb = pos + (size_t)b * N_ * 3;
  const float px = pb[n * 3 + 0];
  const float py = pb[n * 3 + 1];
  const float pz = pb[n * 3 + 2];

  __shared__ float sp[2][TC * 3];
  // low 32 bits of the generic pointer == LDS byte offset (aperture scheme)
  const unsigned lds_base = (unsigned)(uintptr_t)(&sp[0][0]);

  float bd[K_]; int bi[K_];
#pragma unroll
  for (int t = 0; t < K_; ++t) { bd[t] = INFINITY; bi[t] = 0; }

  const int m_begin = seg * SEGLEN;

  // Prologue: DMA tile 0 into buffer 0.
  if (wave == 0) {
    tdm_load_1d(lds_base, pb + (size_t)m_begin * 3, TC * 3);
    __builtin_amdgcn_s_wait_tensorcnt(0);
  }
  __syncthreads();

  for (int c = 0; c < NCHUNK; ++c) {
    const int m0 = m_begin + c * TC;
    const float* cur = sp[c & 1];

    // Kick off DMA for the next tile into the other buffer (overlaps compute).
    if (wave == 0 && (c + 1) < NCHUNK)
      tdm_load_1d(lds_base + (unsigned)(((c + 1) & 1) * (TC * 3 * 4)),
                  pb + (size_t)(m0 + TC) * 3, TC * 3);

#pragma unroll 4
    for (int mm = 0; mm < TC; ++mm) {
      const float dx = px - cur[mm * 3 + 0];
      const float dy = py - cur[mm * 3 + 1];
      const float dz = pz - cur[mm * 3 + 2];
      const float d  = dx * dx + dy * dy + dz * dz;
      if (d < bd[K_ - 1]) {      // threshold test: rarely true after warm-up
        float cd = d; int ci = m0 + mm;
#pragma unroll
        for (int t = 0; t < K_; ++t) {
          if (cd < bd[t]) {
            float td = bd[t]; int ti = bi[t];
            bd[t] = cd; bi[t] = ci; cd = td; ci = ti;
          }
        }
      }
    }

    __syncthreads();                                   // all waves done with buf c
    if (wave == 0 && (c + 1) < NCHUNK)
      __builtin_amdgcn_s_wait_tensorcnt(0);            // next buffer landed
    __syncthreads();                                   // release next tile
  }

  const size_t o = (((size_t)b * N_ + n) * SEG + seg) * K_;
#pragma unroll
  for (int t = 0; t < K_; ++t) { pd[o + t] = bd[t]; pi[o + t] = bi[t]; }
}

// ---------------------------------------------------------------------------
// Phase 2 KNN: merge SEG sorted partial lists -> final top-16 indices.
// Each partial list is ascending, so we can early-exit per list.
// ---------------------------------------------------------------------------
__global__ __launch_bounds__(256) void knn_merge(const float* __restrict__ pd,
                                                 const int*   __restrict__ pi,
                                                 int*         __restrict__ idx) {
  const int q = blockIdx.x * 256 + threadIdx.x;   // [0, B*N)
  const size_t base = (size_t)q * SEG * K_;

  float bd[K_]; int bi[K_];
#pragma unroll
  for (int t = 0; t < K_; ++t) { bd[t] = pd[base + t]; bi[t] = pi[base + t]; }

  for (int s = 1; s < SEG; ++s) {
    const size_t o = base + (size_t)s * K_;
    for (int e = 0; e < K_; ++e) {
      const float d = pd[o + e];
      if (!(d < bd[K_ - 1])) break;   // ascending list: rest cannot qualify
      float cd = d; int ci = pi[o + e];
#pragma unroll
      for (int t = 0; t < K_; ++t) {
        if (cd < bd[t]) {
          float td = bd[t]; int ti = bi[t];
          bd[t] = cd; bi[t] = ci; cd = td; ci = ti;
        }
      }
    }
  }
#pragma unroll
  for (int t = 0; t < K_; ++t) idx[(size_t)q * K_ + t] = bi[t];
}

// ---------------------------------------------------------------------------
// h = x @ W^T via V_WMMA_F32_16X16X4_F32.
// One 16x16 output tile per wave; K loop over 256 in steps of 4.
// A layout (16x4 f32): lane<16 holds M=lane, K={0,1}; lane>=16 holds M=lane-16,
// K={2,3}. B (4x16) mirrors this with N=lane&15. C/D: VGPR r -> row r
// (lanes 0-15) / row r+8 (lanes 16-31).
// ---------------------------------------------------------------------------
__global__ __launch_bounds__(256) void gemm_wmma(const float* __restrict__ x,
                                                 const float* __restrict__ W,
                                                 float* __restrict__ h) {
  const int lane = threadIdx.x & 31;
  const int wave = threadIdx.x >> 5;
  const int tile = blockIdx.x * 8 + wave;        // 16384 tiles total
  const int row0 = (tile >> 4) * 16;             // 1024 row tiles
  const int d0   = (tile & 15) * 16;             // 16 col tiles
  const int r    = lane & 15;                    // A row / B col for this lane
  const int hi   = lane >> 4;
  const int kLo  = hi * 2;
  const int hi8  = hi * 8;

  const float* xr = x + (size_t)(row0 + r) * D_ + kLo;   // A: x[row, k]
  const float* wr = W + (size_t)(d0 + r) * D_ + kLo;     // B: W[col, k] (=> W^T)

  v8f acc = {};
#pragma unroll 8
  for (int k0 = 0; k0 < D_; k0 += 4) {
    const v2f a  = *(const v2f*)(xr + k0);   // 8B-aligned (k even)
    const v2f bb = *(const v2f*)(wr + k0);
    acc = __builtin_amdgcn_wmma_f32_16x16x4_f32(
        /*neg_a=*/false, a, /*neg_b=*/false, bb,
        /*c_mod=*/(short)0, acc, /*reuse_a=*/false, /*reuse_b=*/false);
  }

  float* hout = h + (size_t)(row0 + hi8) * D_ + d0 + r;
#pragma unroll
  for (int t = 0; t < 8; ++t) hout[(size_t)t * D_] = acc[t];
}

// ---------------------------------------------------------------------------
// GAT aggregation: one wave per query. Lane l owns channels [8l, 8l+8) which
// lie entirely inside head (l>>3). Per-head dots reduced with shfl_xor 1/2/4.
// Single pass over 16 neighbors with online softmax (leaky-relu'd scores),
// then residual + ReLU. Neighbor gathers stay in the 192 MB L2 (h = 16.8 MB).
// ---------------------------------------------------------------------------
__global__ __launch_bounds__(256) void gat_agg(const float* __restrict__ h,
                                               const float* __restrict__ x,
                                               const float* __restrict__ att,
                                               const int*   __restrict__ idx,
                                               float* __restrict__ out) {
  const int lane = threadIdx.x & 31;
  const int wave = threadIdx.x >> 5;
  const int q    = blockIdx.x * 8 + wave;       // [0, B*N)
  const int b    = q >> 13;                     // N = 8192
  const int head = lane >> 3;
  const int c0   = lane * 8;                    // global channel base
  const int hch  = (lane & 7) * 8;              // channel within head

  float as[8], an[8];
#pragma unroll
  for (int j = 0; j < 8; ++j) {
    as[j] = att[head * (2 * HD_) + hch + j];
    an[j] = att[head * (2 * HD_) + HD_ + hch + j];
  }

  // self score
  const float* hq = h + (size_t)q * D_ + c0;
  const float4 hv0 = *(const float4*)(hq);
  const float4 hv1 = *(const float4*)(hq + 4);
  float ps = hv0.x * as[0] + hv0.y * as[1] + hv0.z * as[2] + hv0.w * as[3]
           + hv1.x * as[4] + hv1.y * as[5] + hv1.z * as[6] + hv1.w * as[7];
  ps += __shfl_xor(ps, 1, 32);
  ps += __shfl_xor(ps, 2, 32);
  ps += __shfl_xor(ps, 4, 32);
  const float s_self = ps;

  const int myidx = idx[(size_t)q * K_ + (lane & (K_ - 1))];

  float m_run = -INFINITY, l_run = 0.f;
  float acc[8];
#pragma unroll
  for (int j = 0; j < 8; ++j) acc[j] = 0.f;

  const float* hb = h + (size_t)b * N_ * D_;
#pragma unroll 1
  for (int k = 0; k < K_; ++k) {
    const int nb = __shfl(myidx, k, 32);
    const float* hn = hb + (size_t)nb * D_ + c0;       // L2-resident gather
    const float4 n0 = *(const float4*)(hn);
    const float4 n1 = *(const float4*)(hn + 4);
    float p = n0.x * an[0] + n0.y * an[1] + n0.z * an[2] + n0.w * an[3]
            + n1.x * an[4] + n1.y * an[5] + n1.z * an[6] + n1.w * an[7];
    p += __shfl_xor(p, 1, 32);
    p += __shfl_xor(p, 2, 32);
    p += __shfl_xor(p, 4, 32);

    const float raw = s_self + p;
    const float lr  = raw >= 0.f ? raw : 0.2f * raw;   // leaky relu
    const float mn  = fmaxf(m_run, lr);
    const float sc  = __expf(m_run - mn);              // 0 on first iter
    const float w   = __expf(lr - mn);
    l_run = l_run * sc + w;
    acc[0] = acc[0] * sc + w * n0.x;  acc[1] = acc[1] * sc + w * n0.y;
    acc[2] = acc[2] * sc + w * n0.z;  acc[3] = acc[3] * sc + w * n0.w;
    acc[4] = acc[4] * sc + w * n1.x;  acc[5] = acc[5] * sc + w * n1.y;
    acc[6] = acc[6] * sc + w * n1.z;  acc[7] = acc[7] * sc + w * n1.w;
    m_run = mn;
  }

  const float inv = 1.f / l_run;
  const float* xq = x + (size_t)q * D_ + c0;
  float* oq = out + (size_t)q * D_ + c0;
#pragma unroll
  for (int j = 0; j < 8; ++j) {
    const float v = acc[j] * inv + xq[j];
    oq[j] = fmaxf(v, 0.f);
  }
}

// ---------------------------------------------------------------------------
extern "C" void kernel_launch(void* const* d_in, const int* in_sizes, int n_in,
                              void* d_out, int out_size, void* d_ws, size_t ws_size,
                              hipStream_t stream) {
  const float* x   = (const float*)d_in[0];   // (B,N,D)
  const float* pos = (const float*)d_in[1];   // (B,N,3)
  const float* W   = (const float*)d_in[2];   // (D,D)
  const float* att = (const float*)d_in[3];   // (1,H,2*hd)
  float* out = (float*)d_out;

  // Workspace layout (~34.6 MB total)
  char* ws = (char*)d_ws;
  size_t off = 0;
  float* h  = (float*)(ws + off); off += (size_t)B_ * N_ * D_ * sizeof(float);        // 16.78 MB
  int*   id = (int*)  (ws + off); off += (size_t)B_ * N_ * K_ * sizeof(int);          //  1.05 MB
  float* pd = (float*)(ws + off); off += (size_t)B_ * N_ * SEG * K_ * sizeof(float);  //  8.39 MB
  int*   pi = (int*)  (ws + off);                                                     //  8.39 MB

  const dim3 blk(256);
  knn_part <<<dim3(N_ / TQ, SEG, B_),                 blk, 0, stream>>>(pos, pd, pi);
  knn_merge<<<dim3((B_ * N_) / 256),                  blk, 0, stream>>>(pd, pi, id);
  gemm_wmma<<<dim3(((B_ * N_ / 16) * (D_ / 16)) / 8), blk, 0, stream>>>(x, W, h);
  gat_agg  <<<dim3((B_ * N_) / 8),                    blk, 0, stream>>>(h, x, att, id, out);
}